// PatchesKernel3D_34892314313629
// MI455X (gfx1250) — compile-verified
//
#include <hip/hip_runtime.h>
#include <hip/hip_bf16.h>
#include <stdint.h>

// im2col / patch extraction:
//   x:   [B=16, C=3, H=512, W=512] f32
//   out: [B, Ho*Wo, C, K*K] f32 with K=8, S=4, P=0, Ho=Wo=127
//   out[((b*Ho*Wo + oh*Wo + ow)*C + c)*64 + kh*8 + kw] = x[b][c][oh*4+kh][ow*4+kw]
//
// Memory-bound (≈250 MB total @ 23.3 TB/s ≈ 10.7 µs floor). Strategy:
//   * async global->LDS b128 copies (CDNA5 ASYNCcnt path), fully coalesced reads
//   * bank-conflict-free padded LDS (row stride 520 floats)
//   * fully coalesced b128 non-temporal global stores

#define KK   8
#define SS   4
#define B_   16
#define C_   3
#define H_   512
#define W_   512
#define HO   127
#define WO   127
#define ROWF 520                 // padded LDS row stride in floats (512 + 8)
#define ROWB (ROWF * 4)          // 2080 bytes, 16B aligned

typedef float v4f __attribute__((ext_vector_type(4)));

__global__ __launch_bounds__(256)
void patches_im2col_kernel(const float* __restrict__ x, float* __restrict__ out) {
    __shared__ float smem[KK * ROWF];   // 8 padded input rows: 16,640 B

    const int tid = threadIdx.x;
    const int blk = blockIdx.x;         // blk = (b*C + c)*HO + oh
    const int oh  = blk % HO;
    const int bc  = blk / HO;           // b*C + c
    const int c   = bc % C_;
    const int b   = bc / C_;

    // ---------------- Stage 1: async copy 8 rows (ih = oh*4 .. oh*4+7) to LDS
    // 8 rows * 128 float4 = 1024 b128 transfers; 4 per thread, all coalesced.
    const uint64_t plane_base =
        (uint64_t)bc * (uint64_t)(H_ * W_) + (uint64_t)(oh * SS) * W_;
    const uint32_t lds0 = (uint32_t)(uintptr_t)(&smem[0]);  // low 32 bits of a
                                                            // flat shared ptr
                                                            // == LDS byte offset
    #pragma unroll
    for (int it = 0; it < 4; ++it) {
        const int i = tid + it * 256;   // 0..1023
        const int r = i >> 7;           // input row 0..7
        const int q = i & 127;          // float4 index within row
        const uint32_t lds_off = lds0 + (uint32_t)(r * ROWB + q * 16);
        const uint64_t gaddr =
            (uint64_t)(uintptr_t)(x + plane_base + (uint64_t)r * W_ + (uint64_t)q * 4);
        asm volatile("global_load_async_to_lds_b128 %0, %1, off"
                     :
                     : "v"(lds_off), "v"(gaddr)
                     : "memory");
    }
    // Wait for this wave's async LDS writes, then sync the workgroup.
    asm volatile("s_wait_asynccnt 0x0" ::: "memory");
    __syncthreads();

    // ---------------- Stage 2: gather 8x8 patches, stream out
    // One float4 per (ow, k4): k4 = kh*2 + half, covering k = kh*8 + half*4 + j.
    // LDS banks: dword index = kh*520 + ow*4 + half*4 -> bank base
    // (8*kh + 4*half + 4*ow) mod 64: all 16 (kh,half) readers distinct.
    const uint64_t outrow0 =
        ((uint64_t)(b * (HO * WO) + oh * WO) * C_ + (uint64_t)c) * (KK * KK);
    for (int t = tid; t < WO * 16; t += 256) {
        const int ow = t >> 4;
        const int k4 = t & 15;
        const int kh = k4 >> 1;
        const int hf = k4 & 1;
        const v4f v = *reinterpret_cast<const v4f*>(
            &smem[kh * ROWF + ow * SS + hf * 4]);
        v4f* dst = reinterpret_cast<v4f*>(
            out + outrow0 + (uint64_t)ow * (C_ * KK * KK) + (uint64_t)(k4 * 4));
        __builtin_nontemporal_store(v, dst);  // NT: don't thrash L2 with output
    }
}

extern "C" void kernel_launch(void* const* d_in, const int* in_sizes, int n_in,
                              void* d_out, int out_size, void* d_ws, size_t ws_size,
                              hipStream_t stream) {
    (void)in_sizes; (void)n_in; (void)out_size; (void)d_ws; (void)ws_size;
    const float* x = (const float*)d_in[0];
    float* out = (float*)d_out;
    dim3 grid(B_ * C_ * HO);   // 6096 workgroups, one per (b, c, oh)
    patches_im2col_kernel<<<grid, 256, 0, stream>>>(x, out);
}